// GNNModel_10960756540092
// MI455X (gfx1250) — compile-verified
//
#include <hip/hip_runtime.h>
#include <math.h>

// Problem constants (match the reference).
#define DIM    512
#define PNODE  100
#define NSAMP  1024
#define MTOT   1124   // PNODE + NSAMP

typedef __attribute__((ext_vector_type(2))) float v2f;
typedef __attribute__((ext_vector_type(8))) float v8f;

#if defined(__AMDGCN__) && __has_builtin(__builtin_amdgcn_global_load_async_to_lds_b32)
#define USE_ASYNC_LDS 1
#else
#define USE_ASYNC_LDS 0
#endif

#if USE_ASYNC_LDS
typedef __attribute__((address_space(1))) int gas_int;   // global (AS1)
typedef __attribute__((address_space(3))) int las_int;   // LDS    (AS3)
#endif

// ---------------------------------------------------------------------------
// Hin = [proxies ; x]   (row-major [MTOT, DIM])
// ---------------------------------------------------------------------------
__global__ void concat_kernel(const float* __restrict__ proxies,
                              const float* __restrict__ x,
                              float* __restrict__ H) {
    int i = blockIdx.x * blockDim.x + threadIdx.x;
    const int psz = PNODE * DIM;
    const int tot = MTOT * DIM;
    if (i < psz)      H[i] = proxies[i];
    else if (i < tot) H[i] = x[i - psz];
}

// ---------------------------------------------------------------------------
// C[M,512] = A[M,512] @ B[512,512] with V_WMMA_F32_16X16X4_F32.
//
// Block = 256 threads = 8 waves; block tile = 128 rows x 64 cols.
// B k-slices (4x64) are staged in LDS (double-buffered) and shared by all
// 8 waves -> 8x less B traffic from L2.  LDS layout pairs rows kk/kk+1 so a
// B fragment is one ds_load_b64.  Staging uses the gfx1250 async
// global->LDS path (ASYNCcnt + s_wait_asynccnt), overlapping the next
// slice's DMA with the current slice's WMMAs.
//
// f32 16x4 A layout: lanes 0-15 hold (row, K=k..k+1); lanes 16-31 K=k+2..k+3.
// f32 4x16 B layout: mirrored.  f32 16x16 C/D: VGPR r -> rows r / r+8.
// ---------------------------------------------------------------------------
__device__ __forceinline__ void stage_b_slice(const float* __restrict__ B,
                                              float* dst, int colB, int k,
                                              int tid) {
    const int r    = tid >> 6;   // 0..3 : row within k-slice
    const int c    = tid & 63;   // 0..63: col within panel
    const int loff = ((r >> 1) * 128) + (c * 2) + (r & 1);  // paired layout
    const int goff = (k + r) * DIM + colB + c;
#if USE_ASYNC_LDS
    __builtin_amdgcn_global_load_async_to_lds_b32(
        (gas_int*)const_cast<float*>(B + goff),
        (las_int*)(dst + loff),
        0, 0);
#else
    dst[loff] = B[goff];
#endif
}

__device__ __forceinline__ void wait_async_lds() {
#if USE_ASYNC_LDS
#if __has_builtin(__builtin_amdgcn_s_wait_asynccnt)
    __builtin_amdgcn_s_wait_asynccnt(0);
#else
    asm volatile("s_wait_asynccnt 0" ::: "memory");
#endif
#endif
}

__global__ __launch_bounds__(256) void gemm_wmma_f32(const float* __restrict__ A,
                                                     const float* __restrict__ B,
                                                     float* __restrict__ C,
                                                     int M) {
    const int tid  = threadIdx.x;
    const int lane = tid & 31;
    const int wave = tid >> 5;          // 0..7
    const int half = lane >> 4;         // 0: lanes 0-15, 1: lanes 16-31
    const int l15  = lane & 15;
    const int rowB = blockIdx.x * 128 + wave * 16;
    const int colB = blockIdx.y * 64;

    __shared__ __align__(16) float bsh[2][256];   // 2 x (4x64) slices, paired

    int arow = rowB + l15;
    if (arow >= M) arow = M - 1;        // clamp loads; stores are guarded

    v8f acc0 = {}, acc1 = {}, acc2 = {}, acc3 = {};

    stage_b_slice(B, &bsh[0][0], colB, 0, tid);
    wait_async_lds();
    __syncthreads();

    for (int k = 0; k < DIM; k += 4) {
        const int buf = (k >> 2) & 1;
        if (k + 4 < DIM)
            stage_b_slice(B, &bsh[buf ^ 1][0], colB, k + 4, tid);

        v2f a;
        a.x = A[arow * DIM + k + 2 * half];
        a.y = A[arow * DIM + k + 2 * half + 1];

        const float* bs = &bsh[buf][half * 128];
        const v2f b0 = *(const v2f*)&bs[(0 * 16 + l15) * 2];
        const v2f b1 = *(const v2f*)&bs[(1 * 16 + l15) * 2];
        const v2f b2 = *(const v2f*)&bs[(2 * 16 + l15) * 2];
        const v2f b3 = *(const v2f*)&bs[(3 * 16 + l15) * 2];

        acc0 = __builtin_amdgcn_wmma_f32_16x16x4_f32(false, a, false, b0,
                                                     (short)0, acc0, false, false);
        acc1 = __builtin_amdgcn_wmma_f32_16x16x4_f32(false, a, false, b1,
                                                     (short)0, acc1, false, false);
        acc2 = __builtin_amdgcn_wmma_f32_16x16x4_f32(false, a, false, b2,
                                                     (short)0, acc2, false, false);
        acc3 = __builtin_amdgcn_wmma_f32_16x16x4_f32(false, a, false, b3,
                                                     (short)0, acc3, false, false);

        wait_async_lds();        // after the WMMAs: DMA overlapped compute
        __syncthreads();
    }

    const int col0 = colB + l15;
#pragma unroll
    for (int r = 0; r < 8; ++r) {
        const int row = rowB + r + 8 * half;
        if (row < M) {
            float* c = C + row * DIM;
            c[col0]      = acc0[r];
            c[col0 + 16] = acc1[r];
            c[col0 + 32] = acc2[r];
            c[col0 + 48] = acc3[r];
        }
    }
}

// ---------------------------------------------------------------------------
// ls[row] = H[row,:] . a_src ;  ld[row] = H[row,:] . a_dst
// ---------------------------------------------------------------------------
__global__ __launch_bounds__(128) void row_dots(const float* __restrict__ H,
                                                const float* __restrict__ a_src,
                                                const float* __restrict__ a_dst,
                                                float* __restrict__ ls,
                                                float* __restrict__ ld) {
    const int row = blockIdx.x;
    const int tid = threadIdx.x;
    const float* h = H + row * DIM;

    float s1 = 0.f, s2 = 0.f;
    for (int i = tid; i < DIM; i += 128) {
        const float v = h[i];
        s1 += v * a_src[i];
        s2 += v * a_dst[i];
    }
    __shared__ float r1[128], r2[128];
    r1[tid] = s1; r2[tid] = s2;
    __syncthreads();
    for (int off = 64; off > 0; off >>= 1) {
        if (tid < off) { r1[tid] += r1[tid + off]; r2[tid] += r2[tid + off]; }
        __syncthreads();
    }
    if (tid == 0) { ls[row] = r1[0]; ld[row] = r2[0]; }
}

__device__ __forceinline__ float lrelu02(float e) { return e > 0.f ? e : 0.2f * e; }

// ---------------------------------------------------------------------------
// Sample aggregation: node = PNODE + blockIdx.x receives from all 100 proxies
// + self.  Closed-form segment softmax, then Out = relu(agg + bias).
// ---------------------------------------------------------------------------
__global__ __launch_bounds__(128) void agg_samples(const float* __restrict__ H,
                                                   const float* __restrict__ ls,
                                                   const float* __restrict__ ld,
                                                   const float* __restrict__ bias,
                                                   float* __restrict__ Out) {
    const int tid  = threadIdx.x;
    const int node = PNODE + blockIdx.x;

    __shared__ float wsh[PNODE];
    __shared__ float red[128];

    const float ldv = ld[node];

    float lmax = -3.4e38f;
    for (int p = tid; p < PNODE; p += 128) {
        const float e = lrelu02(ls[p] + ldv);
        wsh[p] = e;
        lmax = fmaxf(lmax, e);
    }
    const float eself = lrelu02(ls[node] + ldv);
    lmax = fmaxf(lmax, eself);

    red[tid] = lmax; __syncthreads();
    for (int off = 64; off > 0; off >>= 1) {
        if (tid < off) red[tid] = fmaxf(red[tid], red[tid + off]);
        __syncthreads();
    }
    const float m = red[0];
    __syncthreads();

    float lsum = 0.f;
    for (int p = tid; p < PNODE; p += 128) {
        const float w = expf(wsh[p] - m);
        wsh[p] = w;
        lsum += w;
    }
    red[tid] = lsum; __syncthreads();
    for (int off = 64; off > 0; off >>= 1) {
        if (tid < off) red[tid] += red[tid + off];
        __syncthreads();
    }
    const float wself = expf(eself - m);
    const float invZ  = 1.f / (red[0] + wself);

    for (int d = tid; d < DIM; d += 128) {
        float acc = wself * H[node * DIM + d];
        for (int p = 0; p < PNODE; ++p)
            acc += wsh[p] * H[p * DIM + d];
        const float o = acc * invZ + bias[d];
        Out[node * DIM + d] = o > 0.f ? o : 0.f;
    }
}

// ---------------------------------------------------------------------------
// Proxy aggregation: node = blockIdx.x receives from all 1024 samples + self.
// ---------------------------------------------------------------------------
__global__ __launch_bounds__(256) void agg_proxies(const float* __restrict__ H,
                                                   const float* __restrict__ ls,
                                                   const float* __restrict__ ld,
                                                   const float* __restrict__ bias,
                                                   float* __restrict__ Out) {
    const int tid  = threadIdx.x;
    const int node = blockIdx.x;

    __shared__ float wsh[NSAMP];
    __shared__ float red[256];

    const float ldv = ld[node];

    float lmax = -3.4e38f;
    for (int s = tid; s < NSAMP; s += 256) {
        const float e = lrelu02(ls[PNODE + s] + ldv);
        wsh[s] = e;
        lmax = fmaxf(lmax, e);
    }
    const float eself = lrelu02(ls[node] + ldv);
    lmax = fmaxf(lmax, eself);

    red[tid] = lmax; __syncthreads();
    for (int off = 128; off > 0; off >>= 1) {
        if (tid < off) red[tid] = fmaxf(red[tid], red[tid + off]);
        __syncthreads();
    }
    const float m = red[0];
    __syncthreads();

    float lsum = 0.f;
    for (int s = tid; s < NSAMP; s += 256) {
        const float w = expf(wsh[s] - m);
        wsh[s] = w;
        lsum += w;
    }
    red[tid] = lsum; __syncthreads();
    for (int off = 128; off > 0; off >>= 1) {
        if (tid < off) red[tid] += red[tid + off];
        __syncthreads();
    }
    const float wself = expf(eself - m);
    const float invZ  = 1.f / (red[0] + wself);

    for (int d = tid; d < DIM; d += 256) {
        float acc = wself * H[node * DIM + d];
        for (int s = 0; s < NSAMP; ++s)
            acc += wsh[s] * H[(PNODE + s) * DIM + d];
        const float o = acc * invZ + bias[d];
        Out[node * DIM + d] = o > 0.f ? o : 0.f;
    }
}

// ---------------------------------------------------------------------------
// preds = H2[P:,:] @ fc_w + fc_b  and copy h = H2[P:,:].
// ---------------------------------------------------------------------------
__global__ __launch_bounds__(128) void fc_out(const float* __restrict__ H2,
                                              const float* __restrict__ fcw,
                                              const float* __restrict__ fcb,
                                              float* __restrict__ out) {
    const int s = blockIdx.x;
    const float* h = H2 + (PNODE + s) * DIM;

    for (int j = threadIdx.x; j < 100; j += 128) {
        float acc = fcb[j];
        for (int k = 0; k < DIM; ++k)
            acc += h[k] * fcw[k * 100 + j];
        out[s * 100 + j] = acc;
    }
    float* oh = out + NSAMP * 100;
    for (int d = threadIdx.x; d < DIM; d += 128)
        oh[s * DIM + d] = h[d];
}

// ---------------------------------------------------------------------------
extern "C" void kernel_launch(void* const* d_in, const int* in_sizes, int n_in,
                              void* d_out, int out_size, void* d_ws, size_t ws_size,
                              hipStream_t stream) {
    const float* x       = (const float*)d_in[0];
    const float* proxies = (const float*)d_in[1];
    const float* W1      = (const float*)d_in[2];
    const float* a_src1  = (const float*)d_in[3];
    const float* a_dst1  = (const float*)d_in[4];
    const float* b1      = (const float*)d_in[5];
    const float* W2      = (const float*)d_in[6];
    const float* a_src2  = (const float*)d_in[7];
    const float* a_dst2  = (const float*)d_in[8];
    const float* b2      = (const float*)d_in[9];
    const float* fc_w    = (const float*)d_in[10];
    const float* fc_b    = (const float*)d_in[11];
    // d_in[12] = src, d_in[13] = dst : dense-bipartite topology is closed-form.

    float* out = (float*)d_out;

    const size_t act = (size_t)MTOT * DIM;
    float* ws0 = (float*)d_ws;          // activation buffer A
    float* ws1 = ws0 + act;             // activation buffer B
    float* ws2 = ws1 + act;             // activation buffer C
    float* ls  = ws2 + act;             // [MTOT]
    float* ld  = ls + 1152;             // [MTOT] (padded)

    const dim3 gGemm((MTOT + 127) / 128, DIM / 64);

    // Hin = [proxies ; x]
    concat_kernel<<<(MTOT * DIM + 255) / 256, 256, 0, stream>>>(proxies, x, ws0);

    // ---- Layer 1 ----
    gemm_wmma_f32<<<gGemm, 256, 0, stream>>>(ws0, W1, ws1, MTOT);
    row_dots<<<MTOT, 128, 0, stream>>>(ws1, a_src1, a_dst1, ls, ld);
    agg_samples<<<NSAMP, 128, 0, stream>>>(ws1, ls, ld, b1, ws2);
    agg_proxies<<<PNODE, 256, 0, stream>>>(ws1, ls, ld, b1, ws2);

    // ---- Layer 2 ----
    gemm_wmma_f32<<<gGemm, 256, 0, stream>>>(ws2, W2, ws0, MTOT);
    row_dots<<<MTOT, 128, 0, stream>>>(ws0, a_src2, a_dst2, ls, ld);
    agg_samples<<<NSAMP, 128, 0, stream>>>(ws0, ls, ld, b2, ws1);
    agg_proxies<<<PNODE, 256, 0, stream>>>(ws0, ls, ld, b2, ws1);

    // ---- Head: preds + h copy ----
    fc_out<<<NSAMP, 128, 0, stream>>>(ws1, fc_w, fc_b, out);
}